// dynFilter_44951127720034
// MI455X (gfx1250) — compile-verified
//
#include <hip/hip_runtime.h>
#include <math.h>

typedef __attribute__((ext_vector_type(2))) float v2f;
typedef __attribute__((ext_vector_type(4))) float v4f;
typedef __attribute__((ext_vector_type(8))) float v8f;

#define KW    5
#define PADW  2
#define CCH   4      // input channels
#define NCH   100    // C*K*K
#define HH    256
#define WW    256

#if __has_builtin(__builtin_amdgcn_tanhf)
__device__ __forceinline__ float fast_tanh(float v) { return __builtin_amdgcn_tanhf(v); }
#else
__device__ __forceinline__ float fast_tanh(float v) { return tanhf(v); }
#endif

// One block = one output row (b,h): 512 threads = 16 waves, each wave owns a
// 16-pixel tile along W. Each lane VALU-accumulates 50 of the 100 products
// into the two A-matrix K-slots its half-wave owns, then a single
// V_WMMA_F32_16X16X4_F32 against an all-ones B reduces the 4 partials across
// lanes: D[m][n] = sum_k A[m][k] for every n. filt (210 MB, read-once) streams
// with non-temporal loads so the 8.4 MB x tensor (25x reuse) stays cached.
__global__ __launch_bounds__(512)
void dynfilter_wmma_kernel(const float* __restrict__ x,
                           const float* __restrict__ filt,
                           float* __restrict__ out)
{
    const int bh   = blockIdx.x;       // b*H + h
    const int b    = bh >> 8;          // / 256
    const int h    = bh & 255;         // % 256
    const int tid  = (int)threadIdx.x;
    const int wave = tid >> 5;
    const int lane = tid & 31;
    const int half = lane >> 4;        // selects A-matrix K-slots {0,1} vs {2,3}
    const int L    = lane & 15;        // pixel within the wave's 16-pixel tile
    const int w    = wave * 16 + L;

    const float* __restrict__ xb = x + (size_t)b * CCH * HH * WW;
    // filt[b, ch, h, w]: consecutive lanes -> consecutive w -> coalesced per channel
    const float* __restrict__ fb = filt + (size_t)b * NCH * HH * WW
                                        + (size_t)h * WW + w;

    float acc0 = 0.f, acc1 = 0.f;
    const int chBase = 2 * half;       // half 0: channels =0,1 mod 4; half 1: =2,3 mod 4

    #pragma unroll
    for (int t = 0; t < 25; ++t) {
        #pragma unroll
        for (int j = 0; j < 2; ++j) {
            const int ch = 4 * t + chBase + j;      // 0..99
            const int c  = ch / 25;                 // ch = c*25 + p
            const int p  = ch - c * 25;             // p  = ky*5 + kx
            const int ky = p / 5;
            const int kx = p - ky * 5;
            const int yy = h + ky - PADW;
            const int xx = w + kx - PADW;

            const float f = __builtin_nontemporal_load(&fb[(size_t)ch * HH * WW]);
            float xv = 0.f;
            if ((unsigned)yy < (unsigned)HH && (unsigned)xx < (unsigned)WW)
                xv = xb[((size_t)c * HH + yy) * WW + xx];

            if (j == 0) acc0 = fmaf(f, xv, acc0);
            else        acc1 = fmaf(f, xv, acc1);
        }
    }

    // A-matrix 16x4 f32 layout (ISA 7.12.2): lanes 0-15 hold M=L, VGPR0=K0/VGPR1=K1;
    // lanes 16-31 hold M=L, VGPR0=K2/VGPR1=K3. B = ones -> every D column = row sum.
    v2f a;    a.x = acc0; a.y = acc1;
    v2f bons; bons.x = 1.0f; bons.y = 1.0f;
    v8f cin = {};
    v8f d = __builtin_amdgcn_wmma_f32_16x16x4_f32(
        /*neg_a=*/false, a, /*neg_b=*/false, bons,
        /*c_mod=*/(short)0, cin, /*reuse_a=*/false, /*reuse_b=*/false);

    // C/D layout: VGPR r -> (M=r, N=lane) lanes 0-15; (M=8+r, N=lane-16) lanes 16-31.
    // Every column is identical, so d[0..7] is valid in every lane: apply hardware
    // tanh unconditionally (8x v_tanh_f32, no exec branching), store from N=0 lanes.
    v4f lo, hi;
    lo.x = fast_tanh(d[0]); lo.y = fast_tanh(d[1]);
    lo.z = fast_tanh(d[2]); lo.w = fast_tanh(d[3]);
    hi.x = fast_tanh(d[4]); hi.y = fast_tanh(d[5]);
    hi.z = fast_tanh(d[6]); hi.w = fast_tanh(d[7]);

    if (L == 0) {   // lane 0 -> pixels wbase+0..7, lane 16 -> wbase+8..15
        float* o = out + (size_t)bh * WW + wave * 16 + half * 8;
        __builtin_nontemporal_store(lo, (v4f*)(o + 0));
        __builtin_nontemporal_store(hi, (v4f*)(o + 4));
    }
}

extern "C" void kernel_launch(void* const* d_in, const int* in_sizes, int n_in,
                              void* d_out, int out_size, void* d_ws, size_t ws_size,
                              hipStream_t stream) {
    const float* x    = (const float*)d_in[0];   // [B, 4, 256, 256] f32
    const float* filt = (const float*)d_in[1];   // [B, 100, 256, 256] f32
    float* out        = (float*)d_out;           // [B, 1, 256, 256] f32

    const int B = in_sizes[0] / (CCH * HH * WW); // 8 for the reference shapes
    dim3 grid(B * HH);   // one block per output row
    dim3 block(512);     // 16 wave32 -> 16 x 16-pixel tiles = 256 = W
    hipLaunchKernelGGL(dynfilter_wmma_kernel, grid, block, 0, stream, x, filt, out);
}